// GraphClusterReshape_62629213110354
// MI455X (gfx1250) — compile-verified
//
#include <hip/hip_runtime.h>

// GraphClusterReshape: out[m, k*64 .. k*64+63] = (nidx[m,k] >= 0) ? features[nidx[m,k], :] : 0
//
// features: [100000, 64] f32   (25.6 MB -> L2-resident on a 192 MB L2)
// nidx:     [50000, 32] i32    (-1 marks invalid neighbour)
// out:      [50000, 2048] f32  (409.6 MB streaming write)
//
// Pure bandwidth problem: ~442 MB HBM traffic -> ~19 us floor at 23.3 TB/s.
// Mapping: a 16-lane group owns 4 consecutive (m,k) pairs ("quad").
//   - 1x global_load_b128 pulls all 4 neighbour indices (shared by the group)
//   - 4x independent global_load_b128 gathers (clustered -> 4 loads in flight
//     per wait, 2 KB of MLP per wave instead of 512 B)
//   - 4x global_store_b128 th:TH_STORE_NT covering 1 KB contiguous output,
//     NT so the write stream doesn't evict the hot feature table from L2.

typedef float v4f __attribute__((ext_vector_type(4)));
typedef int   v4i __attribute__((ext_vector_type(4)));

#define FEAT_V4 16  // 64 floats per row = 16 x float4

__global__ __launch_bounds__(256) void graph_gather_quad_kernel(
    const float* __restrict__ feat,
    const int*   __restrict__ nidx,
    float*       __restrict__ out,
    unsigned total_quads)                 // = (M*K)/4 = 400,000
{
    const unsigned gid    = blockIdx.x * 256u + threadIdx.x;
    const unsigned lane16 = gid & 15u;    // which float4 within the 64-float row
    const unsigned quad   = gid >> 4;     // owns pairs [quad*4, quad*4+4)
    if (quad >= total_quads) return;

    // One b128 fetches 4 neighbour indices; all 16 lanes of the group read
    // the same 16 B line -> hits WGP$/L0.
    const v4i idx4 = ((const v4i*)nidx)[quad];
    const int i0 = idx4.x, i1 = idx4.y, i2 = idx4.z, i3 = idx4.w;

    // Branchless clamp: always issue the gather (row 0 is valid memory).
    const v4f* __restrict__ src = (const v4f*)feat;
    const size_t col = lane16;
    v4f a = src[(size_t)(i0 < 0 ? 0 : i0) * FEAT_V4 + col];
    v4f b = src[(size_t)(i1 < 0 ? 0 : i1) * FEAT_V4 + col];
    v4f c = src[(size_t)(i2 < 0 ? 0 : i2) * FEAT_V4 + col];
    v4f d = src[(size_t)(i3 < 0 ? 0 : i3) * FEAT_V4 + col];

    const v4f z = (v4f){0.0f, 0.0f, 0.0f, 0.0f};
    if (i0 < 0) a = z;
    if (i1 < 0) b = z;
    if (i2 < 0) c = z;
    if (i3 < 0) d = z;

    // 4 pairs x 256 B = 1 KB contiguous per group.
    v4f* dst = (v4f*)out + (size_t)quad * (4 * FEAT_V4) + lane16;
    __builtin_nontemporal_store(a, dst);
    __builtin_nontemporal_store(b, dst + FEAT_V4);
    __builtin_nontemporal_store(c, dst + 2 * FEAT_V4);
    __builtin_nontemporal_store(d, dst + 3 * FEAT_V4);
}

extern "C" void kernel_launch(void* const* d_in, const int* in_sizes, int n_in,
                              void* d_out, int out_size, void* d_ws, size_t ws_size,
                              hipStream_t stream) {
    (void)n_in; (void)out_size; (void)d_ws; (void)ws_size;

    const float* feat = (const float*)d_in[0];   // [N_NODES, 64] f32
    const int*   nidx = (const int*)d_in[1];     // [M, K] i32
    float*       out  = (float*)d_out;           // [M, K*64] f32

    const unsigned total_pairs = (unsigned)in_sizes[1];  // M*K (1,600,000; /4 exact)
    const unsigned total_quads = total_pairs / 4u;
    const unsigned long long threads = (unsigned long long)total_quads * 16ull;
    const unsigned block = 256u;
    const unsigned grid  = (unsigned)((threads + block - 1) / block);

    graph_gather_quad_kernel<<<grid, block, 0, stream>>>(feat, nidx, out, total_quads);
}